// ModelNew_68161130987805
// MI455X (gfx1250) — compile-verified
//
#include <hip/hip_runtime.h>
#include <cstdint>

typedef __attribute__((ext_vector_type(2))) float v2f;
typedef __attribute__((ext_vector_type(4))) float v4f;
typedef __attribute__((ext_vector_type(8))) float v8f;
typedef unsigned int u32x4 __attribute__((ext_vector_type(4)));
typedef int          i32x4 __attribute__((ext_vector_type(4)));
typedef int          i32x8 __attribute__((ext_vector_type(8)));

// Problem sizes (fixed by the reference setup_inputs)
static constexpr int BATCH = 16384;
static constexpr int K     = 4096;   // input_size
static constexpr float SCALE = 0.5f;

// TDM availability / arity detection:
//  - amdgpu-toolchain (clang-23 + therock-10.0 headers) ships the TDM header
//    and uses the 6-arg builtin.
//  - ROCm 7.2 (clang-22) has the 5-arg builtin, no header.
#if defined(__HIP_DEVICE_COMPILE__) && __has_builtin(__builtin_amdgcn_tensor_load_to_lds) && \
    __has_builtin(__builtin_amdgcn_s_wait_tensorcnt)
  #define HAVE_TDM 1
  #if __has_include(<hip/amd_detail/amd_gfx1250_TDM.h>)
    #define TDM_SIX_ARGS 1
  #endif
#endif

// ---------------------------------------------------------------------------
// Kernel 0: zero the wsum scratch (d_ws is poisoned by the harness; we must
// re-zero it every call so the atomic accumulation is deterministic).
// ---------------------------------------------------------------------------
__global__ __launch_bounds__(256) void zero_wsum(float* __restrict__ wsum) {
    int i = blockIdx.x * 256 + (threadIdx.x & 255);   // 4096 threads
    wsum[i] = 0.0f;
}

// ---------------------------------------------------------------------------
// Kernel 1: wsum[c] = sum_h weight[h][c]. 256 blocks x 256 threads.
// blockIdx.x = {rowSplit[4b], colBlock[4b]}; each block reduces a 256x256
// panel with fully coalesced reads, then atomically folds into wsum.
// 64 MB streamed once -> ~2.7 us at 23.3 TB/s.
// ---------------------------------------------------------------------------
__global__ __launch_bounds__(256) void weight_colsum(const float* __restrict__ w,
                                                     float* __restrict__ wsum) {
    const int tid = threadIdx.x & 255;
    const int cb  = blockIdx.x & 15;          // column block (16 x 256 cols)
    const int rs  = blockIdx.x >> 4;          // row split    (16 x 256 rows)
    const int col = cb * 256 + tid;
    const float* p = w + (size_t)rs * 256 * K + col;
    float s = 0.0f;
#pragma unroll 8
    for (int r = 0; r < 256; ++r) {
        s += p[(size_t)r * K];
    }
    atomicAdd(&wsum[col], s);                 // global_atomic_add_f32
}

// ---------------------------------------------------------------------------
// Kernel 2: out[b] = SCALE * dot(x[b], wsum) via V_WMMA_F32_16X16X4_F32.
//
// Block = 128 threads (4 waves), one 16-row tile of x per block. wsum is
// staged into 16 KB of LDS once per block — via the Tensor Data Mover when
// the toolchain exposes it (tensor_load_to_lds + s_wait_tensorcnt), else a
// register-staged clause of B128 loads. Each wave covers a K-span of 1024.
//
// Per iteration (K advances by 8):
//   lanes 0..15  load x[row, kb..kb+3]   (float4) -> A frags for 2 WMMA steps
//   lanes 16..31 load x[row, kb+4..kb+7] (float4)
//   matching wsum float4 from LDS -> B frags (broadcast across all N columns)
// A and B use the same global-k permutation, so the dot product is exact.
// Since all 16 N-columns of B are equal, every lane's C/D VGPR r holds the
// dot-partial of row (half*8 + r) of the tile.
// ---------------------------------------------------------------------------
__global__ __launch_bounds__(128) void gemv_wmma_f32(const float* __restrict__ x,
                                                     const float* __restrict__ wsum,
                                                     float* __restrict__ out) {
    __shared__ float lds_w[K];
    __shared__ float red[16];

    const int tid = threadIdx.x & 127;        // provably < 128: no exec guards

    if (tid < 16) red[tid] = 0.0f;

#if defined(HAVE_TDM)
    // --- TDM staging: one 2D tile (1 row x 4096 f32) global -> LDS ---------
    if (tid == 0) {
        const uint64_t ga   = (uint64_t)(const void*)wsum;   // byte address
        const uint32_t lbase = (uint32_t)(size_t)(void*)lds_w; // LDS offset (low 32 bits)
        // D# group 0: [1:0]=count=1, [63:32]=lds_addr, [120:64]=global_addr,
        //             [127:126]=type=2
        u32x4 g0 = { 1u,
                     lbase,
                     (uint32_t)ga,
                     (uint32_t)(((ga >> 32) & 0x01FFFFFFull) | (2u << 30)) };
        // D# group 1: [17:16]=data_size=2 (4B), [79:48]=tensor_dim0=4096,
        //             [111:80]=tensor_dim1=1, [127:112]=tile_dim0=4096,
        //             [143:128]=tile_dim1=1, [207:160]=tensor_dim0_stride=4096
        i32x8 g1 = { (int)(2u << 16),          // data_size = 4 bytes
                     (int)(0x1000u << 16),     // tensor_dim0[15:0] = 4096
                     (int)(1u << 16),          // tensor_dim1 = 1
                     (int)(0x1000u << 16),     // tile_dim0 = 4096
                     1,                        // tile_dim1 = 1
                     4096,                     // tensor_dim0_stride[31:0]
                     0, 0 };
        i32x4 gz = { 0, 0, 0, 0 };
  #if defined(TDM_SIX_ARGS)
        i32x8 gz8 = { 0, 0, 0, 0, 0, 0, 0, 0 };
        __builtin_amdgcn_tensor_load_to_lds(g0, g1, gz, gz, gz8, 0);
  #else
        __builtin_amdgcn_tensor_load_to_lds(g0, g1, gz, gz, 0);
  #endif
        __builtin_amdgcn_s_wait_tensorcnt(0); // TENSORcnt==0: tile is in LDS
    }
#else
    // --- Fallback staging: clause of 8 B128 loads, one wait, 8 DS stores ---
    {
        const v4f* src = (const v4f*)wsum;
        v4f*       dst = (v4f*)lds_w;
        v4f tmp[8];
#pragma unroll
        for (int j = 0; j < 8; ++j) tmp[j] = src[tid + j * 128];
#pragma unroll
        for (int j = 0; j < 8; ++j) dst[tid + j * 128] = tmp[j];
    }
#endif
    __syncthreads();

    const int wave = tid >> 5;                 // 0..3
    const int lane = tid & 31;
    const int half = lane >> 4;                // 0: lanes 0-15, 1: lanes 16-31
    const int mrow = lane & 15;                // tile row held by this lane

    const size_t row   = (size_t)blockIdx.x * 16 + mrow;
    const float* xrow  = x + row * (size_t)K;
    const int    kbeg  = wave * (K / 4);       // 1024-wide span per wave
    const int    kend  = kbeg + (K / 4);
    const int    kofs  = half * 4;             // upper half-wave carries kb+4..kb+7

    v8f acc = {};
#pragma unroll 4
    for (int kb = kbeg; kb < kend; kb += 8) {
        const int kl = kb + kofs;
        v4f a4 = *(const v4f*)(xrow + kl);      // global B128, streams x
        v4f b4 = *(const v4f*)(&lds_w[kl]);     // LDS B128

        v2f a0 = {a4[0], a4[1]};
        v2f a1 = {a4[2], a4[3]};
        v2f b0 = {b4[0], b4[1]};
        v2f b1 = {b4[2], b4[3]};

        // D = A x B + C, full fp32. 8 args:
        // (neg_a, A, neg_b, B, c_mod, C, reuse_a, reuse_b)
        acc = __builtin_amdgcn_wmma_f32_16x16x4_f32(
                  false, a0, false, b0, (short)0, acc, false, false);
        acc = __builtin_amdgcn_wmma_f32_16x16x4_f32(
                  false, a1, false, b1, (short)0, acc, false, false);
    }

    // C/D layout: lanes 0-15 VGPR r == row r (all N columns identical);
    // lanes 16-31 VGPR r == row 8+r. One lane per half-wave folds its 8 rows.
    if ((lane & 15) == 0) {
        const int base = half * 8;
#pragma unroll
        for (int r = 0; r < 8; ++r) {
            atomicAdd(&red[base + r], acc[r]);  // ds_add_f32
        }
    }
    __syncthreads();

    if (tid < 16) {
        out[(size_t)blockIdx.x * 16 + tid] = SCALE * red[tid];
    }
}

// ---------------------------------------------------------------------------
extern "C" void kernel_launch(void* const* d_in, const int* in_sizes, int n_in,
                              void* d_out, int out_size, void* d_ws, size_t ws_size,
                              hipStream_t stream) {
    const float* x    = (const float*)d_in[0];   // [16384, 4096] fp32
    const float* w    = (const float*)d_in[1];   // [4096, 4096] fp32
    float*       out  = (float*)d_out;           // [16384] fp32
    float*       wsum = (float*)d_ws;            // 4096 fp32 scratch

    zero_wsum<<<K / 256, 256, 0, stream>>>(wsum);
    weight_colsum<<<256, 256, 0, stream>>>(w, wsum);
    gemv_wmma_f32<<<BATCH / 16, 128, 0, stream>>>(x, wsum, out);
}